// GraphSAGE_5119601017098
// MI455X (gfx1250) — compile-verified
//
#include <hip/hip_runtime.h>

typedef float v2f __attribute__((ext_vector_type(2)));
typedef float v8f __attribute__((ext_vector_type(8)));

#define SAGE_N 50000
#define SAGE_E 800000

// ---------------------------------------------------------------------------
// Degree counts: cnt[dst] += 1 per edge (computed once; same for all layers).
// ---------------------------------------------------------------------------
__global__ void sage_count(const int* __restrict__ dst, float* __restrict__ cnt, int E) {
    int e = blockIdx.x * blockDim.x + threadIdx.x;
    if (e < E) atomicAdd(&cnt[dst[e]], 1.0f);
}

// ---------------------------------------------------------------------------
// Edge scatter: agg[dst] += x[src]. One wave per edge; each lane moves a
// float4 chunk of the 128-float row (coalesced 512B gather per edge).
// agg (25.6MB) and x (25.6MB) both fit in the 192MB L2 -> atomics resolve in L2.
// ---------------------------------------------------------------------------
__global__ void sage_scatter(const int* __restrict__ src, const int* __restrict__ dst,
                             const float* __restrict__ x, float* __restrict__ agg, int E) {
    long long idx = (long long)blockIdx.x * blockDim.x + threadIdx.x;
    int e = (int)(idx >> 5);   // wave-uniform (blockDim multiple of 32)
    int c = (int)(idx & 31);   // lane: which float4 chunk of the row
    if (e >= E) return;
    int s = src[e];
    int d = dst[e];
    const float4 v = *(const float4*)(x + (long long)s * 128 + c * 4);
    float* ap = agg + (long long)d * 128 + c * 4;
    atomicAdd(ap + 0, v.x);
    atomicAdd(ap + 1, v.y);
    atomicAdd(ap + 2, v.z);
    atomicAdd(ap + 3, v.w);
}

// ---------------------------------------------------------------------------
// Fused SAGE linear layer via V_WMMA_F32_16X16X4_F32 (exact fp32 matrix math):
//   out = act( (agg/max(cnt,1)) @ Wl^T + b + xin @ Wr^T )
// One wave computes one 16x16 output tile. A frag (16x4): lane m=lane&15 is
// row M=m; half=lane>>4 selects K pair {0,1} vs {2,3} -> a float2 load.
// B frag (4x16): B[k][j] = W[j][k], lane m is column j0+m, same K-pair split
// -> also a float2 load from W's row j0+m. C/D: VGPR r holds row r0+r+8*half,
// col j0+m. All dims (50000, 128, 64) are multiples of 16: no masking, EXEC
// stays all-ones inside the wave as WMMA requires.
// ---------------------------------------------------------------------------
__global__ void sage_gemm_wmma(const float* __restrict__ agg,
                               const float* __restrict__ cnt,
                               const float* __restrict__ xin,
                               const float* __restrict__ Wl,
                               const float* __restrict__ bias,
                               const float* __restrict__ Wr,
                               float* __restrict__ out,
                               int K, int Ncols, int nTiles, int doRelu) {
    const int lane = threadIdx.x & 31;
    const int wave = threadIdx.x >> 5;
    const int tile = blockIdx.x * (blockDim.x >> 5) + wave;
    if (tile >= nTiles) return;                  // wave-uniform branch

    const int colTiles = Ncols >> 4;
    const int rt = tile / colTiles;
    const int ct = tile - rt * colTiles;
    const int r0 = rt << 4;
    const int j0 = ct << 4;
    const int m    = lane & 15;
    const int half = lane >> 4;
    const int ko   = 2 * half;                   // K offset within 4-step: {0,1} or {2,3}

    // per-lane mean-normalization scale for row r0+m (folded into A fragment)
    const float inv = 1.0f / fmaxf(cnt[r0 + m], 1.0f);

    v8f acc = {};

    // pass 1: (agg * inv) @ Wl^T
    const float* Arow = agg + (long long)(r0 + m) * K;
    const float* Brow = Wl  + (long long)(j0 + m) * K;
    for (int kk = 0; kk < K; kk += 4) {
        v2f a = *(const v2f*)(Arow + kk + ko);
        a *= inv;
        v2f bf = *(const v2f*)(Brow + kk + ko);
        acc = __builtin_amdgcn_wmma_f32_16x16x4_f32(
            /*neg_a=*/false, a, /*neg_b=*/false, bf,
            /*c_mod=*/(short)0, acc, /*reuse_a=*/false, /*reuse_b=*/false);
    }

    // pass 2: xin @ Wr^T accumulated into the same tile
    const float* Xrow  = xin + (long long)(r0 + m) * K;
    const float* Brow2 = Wr  + (long long)(j0 + m) * K;
    for (int kk = 0; kk < K; kk += 4) {
        v2f a  = *(const v2f*)(Xrow  + kk + ko);
        v2f bf = *(const v2f*)(Brow2 + kk + ko);
        acc = __builtin_amdgcn_wmma_f32_16x16x4_f32(
            false, a, false, bf, (short)0, acc, false, false);
    }

    const float bj = bias[j0 + m];
#pragma unroll
    for (int r = 0; r < 8; ++r) {
        float v = acc[r] + bj;
        if (doRelu) v = fmaxf(v, 0.0f);
        out[(long long)(r0 + r + 8 * half) * Ncols + (j0 + m)] = v;
    }
}

// ---------------------------------------------------------------------------
// Driver: 3 layers. Workspace: agg[N*128] | cnt[N] | h1[N*128] | h2[N*128].
// ---------------------------------------------------------------------------
extern "C" void kernel_launch(void* const* d_in, const int* in_sizes, int n_in,
                              void* d_out, int out_size, void* d_ws, size_t ws_size,
                              hipStream_t stream) {
    const float* x   = (const float*)d_in[0];
    const int*   ei  = (const int*)d_in[1];   // jax default: int64 request -> int32
    const float* W1l = (const float*)d_in[2];
    const float* b1  = (const float*)d_in[3];
    const float* W1r = (const float*)d_in[4];
    const float* W2l = (const float*)d_in[5];
    const float* b2  = (const float*)d_in[6];
    const float* W2r = (const float*)d_in[7];
    const float* W3l = (const float*)d_in[8];
    const float* b3  = (const float*)d_in[9];
    const float* W3r = (const float*)d_in[10];
    float* out = (float*)d_out;

    const int* src = ei;
    const int* dst = ei + SAGE_E;

    float* agg = (float*)d_ws;                         // N*128
    float* cnt = agg + (size_t)SAGE_N * 128;           // N
    float* h1  = cnt + SAGE_N;                         // N*128
    float* h2  = h1 + (size_t)SAGE_N * 128;            // N*128

    const size_t aggBytes = (size_t)SAGE_N * 128 * sizeof(float);

    const int cntBlocks  = (SAGE_E + 255) / 256;
    const int scatBlocks = (int)(((long long)SAGE_E * 32 + 255) / 256);   // 100000
    const int tilesH     = (SAGE_N / 16) * (128 / 16);                    // 25000
    const int tilesO     = (SAGE_N / 16) * (64 / 16);                     // 12500
    const int gemmBlkH   = (tilesH + 7) / 8;   // 8 waves (16x16 tiles) per 256-thr block
    const int gemmBlkO   = (tilesO + 7) / 8;

    // ---- layer 1 ----
    hipMemsetAsync(agg, 0, aggBytes + SAGE_N * sizeof(float), stream);    // agg + cnt
    sage_count<<<cntBlocks, 256, 0, stream>>>(dst, cnt, SAGE_E);
    sage_scatter<<<scatBlocks, 256, 0, stream>>>(src, dst, x, agg, SAGE_E);
    sage_gemm_wmma<<<gemmBlkH, 256, 0, stream>>>(agg, cnt, x, W1l, b1, W1r, h1,
                                                 128, 128, tilesH, 1);
    // ---- layer 2 ----
    hipMemsetAsync(agg, 0, aggBytes, stream);
    sage_scatter<<<scatBlocks, 256, 0, stream>>>(src, dst, h1, agg, SAGE_E);
    sage_gemm_wmma<<<gemmBlkH, 256, 0, stream>>>(agg, cnt, h1, W2l, b2, W2r, h2,
                                                 128, 128, tilesH, 1);
    // ---- layer 3 (no activation, 64 outputs) ----
    hipMemsetAsync(agg, 0, aggBytes, stream);
    sage_scatter<<<scatBlocks, 256, 0, stream>>>(src, dst, h2, agg, SAGE_E);
    sage_gemm_wmma<<<gemmBlkO, 256, 0, stream>>>(agg, cnt, h2, W3l, b3, W3r, out,
                                                 128, 64, tilesO, 0);
}